// FullAttentionBlock_20512763806098
// MI455X (gfx1250) — compile-verified
//
#include <hip/hip_runtime.h>
#include <hip/hip_bf16.h>
#include <math.h>
#include <stdint.h>

// ---------------------------------------------------------------------------
// Problem constants (fixed by the reference)
// ---------------------------------------------------------------------------
#define B_DIM   2
#define S_DIM   2048
#define HID_DIM 1024
#define H_DIM   16
#define HKV_DIM 8
#define D_DIM   128
#define G_DIM   2            // H / HKV
#define EPS_F   1e-6f
#define QSCALE  0.08838834764831845f   // 1/sqrt(128)
#define LN_ROPE 13.815510557964274f    // ln(1e6)

typedef __attribute__((ext_vector_type(2))) float v2f;
typedef __attribute__((ext_vector_type(8))) float v8f;
typedef unsigned int u32;
typedef unsigned long long u64;
typedef __attribute__((ext_vector_type(4))) u32 su4;
typedef __attribute__((ext_vector_type(8))) u32 su8;

// Full-precision fp32 WMMA: D(16x16,f32) = A(16x4,f32) * B(4x16,f32) + C
__device__ __forceinline__ v8f wmma_f32(v2f a, v2f b, v8f c) {
  return __builtin_amdgcn_wmma_f32_16x16x4_f32(
      /*neg_a=*/false, a, /*neg_b=*/false, b,
      /*c_mod=*/(short)0, c, /*reuse_a=*/false, /*reuse_b=*/false);
}

// ---------------------------------------------------------------------------
// Tensor Data Mover: async 2D tile load global -> LDS. Descriptor groups are
// wave-uniform SGPR tuples; D# pad fields implement the bank-conflict-free
// LDS row strides directly (pad_interval code k -> pad after 2^(k+1) DWORDs;
// pad_amount code k -> insert (k+1) DWORDs).
// ---------------------------------------------------------------------------
__device__ __forceinline__ void tdm_load_2d(u32 lds_off, const float* gptr,
                                            u32 tile_d0, u32 tile_d1,
                                            u32 stride_elems,
                                            u32 pad_interval_code,
                                            u32 pad_amount_code) {
  const u64 ga = (u64)(uintptr_t)gptr;
  su4 g0;
  g0.x = 0x1u;                              // count=1 (valid user descriptor)
  g0.y = lds_off;                           // lds_addr (bytes)
  g0.z = (u32)ga;                           // global_addr[31:0]
  g0.w = (u32)(ga >> 32) | (2u << 30);      // global_addr[56:32] | type=2
  su8 g1;
  g1.s0 = (2u << 16)                        // data_size = 4 bytes
        | (1u << 20)                        // pad_enable
        | (pad_interval_code << 22)
        | (pad_amount_code << 25);          // workgroup_mask = 0 (no cluster)
  g1.s1 = (tile_d0 & 0xFFFFu) << 16;        // tensor_dim0[15:0] (== tile_d0)
  g1.s2 = (tile_d0 >> 16) | ((tile_d1 & 0xFFFFu) << 16);  // dim0 hi | dim1 lo
  g1.s3 = (tile_d1 >> 16) | (tile_d0 << 16);               // dim1 hi | tile_dim0
  g1.s4 = tile_d1 & 0xFFFFu;                // tile_dim1 | tile_dim2=0
  g1.s5 = stride_elems;                     // tensor_dim0_stride[31:0]
  g1.s6 = 0u;
  g1.s7 = 0u;
  asm volatile("tensor_load_to_lds %0, %1" :: "s"(g0), "s"(g1) : "memory");
}

// ---------------------------------------------------------------------------
// Kernel 1: row-major fp32 GEMM  C[M,N] = A[M,K] * B[K,N]
//   V_WMMA_F32_16X16X4_F32 compute, TDM double-buffered LDS staging.
//   block = 256 threads (8 waves), block tile 128x128, BK = 16
//   wave tile 64(M) x 32(N); wave 0 drives the TDM pipeline one stage ahead.
// ---------------------------------------------------------------------------
#define GEMM_BM 128
#define GEMM_BN 128
#define GEMM_BK 16
#define AS_STRIDE 20   // 16 data + 4 pad DWORDs per row (TDM pad codes 3/3)
#define BS_STRIDE 136  // 128 data + 8 pad DWORDs per row (TDM pad codes 6/7)

__global__ __launch_bounds__(256)
void gemm_f32_wmma(const float* __restrict__ A, const float* __restrict__ Bm,
                   float* __restrict__ Cm, int Mdim, int Ndim, int Kdim) {
  __shared__ __align__(16) float As[2][GEMM_BM * AS_STRIDE];
  __shared__ __align__(16) float Bs[2][GEMM_BK * BS_STRIDE];

  const int tid  = threadIdx.x;
  const int lane = tid & 31;
  const int wave = tid >> 5;
  const int half = lane >> 4;
  const int lm   = lane & 15;

  const int wm = wave >> 2;       // 0..1  (M direction, 64 rows each)
  const int wn = wave & 3;        // 0..3  (N direction, 32 cols each)

  const int rowBase = blockIdx.y * GEMM_BM;
  const int colBase = blockIdx.x * GEMM_BN;

  v8f acc[4][2];
#pragma unroll
  for (int mi = 0; mi < 4; ++mi)
#pragma unroll
    for (int ni = 0; ni < 2; ++ni)
      acc[mi][ni] = (v8f){0.f, 0.f, 0.f, 0.f, 0.f, 0.f, 0.f, 0.f};

  if (wave == 0) {
    tdm_load_2d((u32)(uintptr_t)&As[0][0], A + (size_t)rowBase * Kdim,
                GEMM_BK, GEMM_BM, (u32)Kdim, 3u, 3u);
    tdm_load_2d((u32)(uintptr_t)&Bs[0][0], Bm + colBase,
                GEMM_BN, GEMM_BK, (u32)Ndim, 6u, 7u);
  }

  for (int kt = 0; kt < Kdim; kt += GEMM_BK) {
    const int buf = (kt >> 4) & 1;

    if (wave == 0) {
      const int kt2 = kt + GEMM_BK;
      if (kt2 < Kdim) {
        tdm_load_2d((u32)(uintptr_t)&As[buf ^ 1][0],
                    A + (size_t)rowBase * Kdim + kt2,
                    GEMM_BK, GEMM_BM, (u32)Kdim, 3u, 3u);
        tdm_load_2d((u32)(uintptr_t)&Bs[buf ^ 1][0],
                    Bm + (size_t)kt2 * Ndim + colBase,
                    GEMM_BN, GEMM_BK, (u32)Ndim, 6u, 7u);
        __builtin_amdgcn_s_wait_tensorcnt(2);  // two older ops (this buf) done
      } else {
        __builtin_amdgcn_s_wait_tensorcnt(0);
      }
    }
    __syncthreads();  // publish TDM-written LDS tile to all waves

    const float* Asb = &As[buf][0];
    const float* Bsb = &Bs[buf][0];
#pragma unroll
    for (int k0 = 0; k0 < GEMM_BK; k0 += 4) {
      v2f af[4], bf[2];
#pragma unroll
      for (int mi = 0; mi < 4; ++mi) {
        const int rr = wm * 64 + mi * 16 + lm;
        af[mi].x = Asb[rr * AS_STRIDE + k0 + 2 * half];
        af[mi].y = Asb[rr * AS_STRIDE + k0 + 2 * half + 1];
      }
#pragma unroll
      for (int ni = 0; ni < 2; ++ni) {
        const int cc = wn * 32 + ni * 16 + lm;
        bf[ni].x = Bsb[(k0 + 2 * half) * BS_STRIDE + cc];
        bf[ni].y = Bsb[(k0 + 2 * half + 1) * BS_STRIDE + cc];
      }
#pragma unroll
      for (int mi = 0; mi < 4; ++mi)
#pragma unroll
        for (int ni = 0; ni < 2; ++ni)
          acc[mi][ni] = wmma_f32(af[mi], bf[ni], acc[mi][ni]);
    }
    __syncthreads();  // all waves done with buf before it is refilled
  }

#pragma unroll
  for (int mi = 0; mi < 4; ++mi)
#pragma unroll
    for (int ni = 0; ni < 2; ++ni) {
      const int col = colBase + wn * 32 + ni * 16 + lm;
#pragma unroll
      for (int r = 0; r < 8; ++r) {
        const int row = rowBase + wm * 64 + mi * 16 + 8 * half + r;
        Cm[(size_t)row * Ndim + col] = acc[mi][ni][r];
      }
    }
}

// ---------------------------------------------------------------------------
// Kernel 2: per-(b,s,head) RMSNorm + RoPE + layout reorder.
// ---------------------------------------------------------------------------
__global__ __launch_bounds__(128)
void norm_rope_reorder(const float* __restrict__ q_raw,  // [B,S,H*D]
                       const float* __restrict__ k_raw,  // [B,S,HKV*D]
                       const float* __restrict__ v_raw,  // [B,S,HKV*D]
                       const float* __restrict__ q_norm_w,
                       const float* __restrict__ k_norm_w,
                       const int*   __restrict__ pos_ids,
                       float* __restrict__ q_proc,       // [B,H,S,D]
                       float* __restrict__ k_out,        // [B,HKV,S,D]
                       float* __restrict__ v_out) {      // [B,HKV,S,D]
  __shared__ float red[D_DIM];
  const int t   = threadIdx.x;
  const int idx = blockIdx.x;
  const int hh  = idx & 31;
  const int s   = (idx >> 5) % S_DIM;
  const int b   = idx / (32 * S_DIM);

  if (hh >= 24) {  // V: plain copy with transpose to [B,HKV,S,D]
    const int vh = hh - 24;
    const float x = v_raw[((size_t)b * S_DIM + s) * (HKV_DIM * D_DIM) + vh * D_DIM + t];
    v_out[(((size_t)b * HKV_DIM + vh) * S_DIM + s) * D_DIM + t] = x;
    return;
  }

  const bool is_q = (hh < H_DIM);
  float x;
  if (is_q)
    x = q_raw[((size_t)b * S_DIM + s) * (H_DIM * D_DIM) + hh * D_DIM + t];
  else
    x = k_raw[((size_t)b * S_DIM + s) * (HKV_DIM * D_DIM) + (hh - 16) * D_DIM + t];

  red[t] = x * x;
  __syncthreads();
#pragma unroll
  for (int st = 64; st > 0; st >>= 1) {
    if (t < st) red[t] += red[t + st];
    __syncthreads();
  }
  const float var = red[0] * (1.0f / (float)D_DIM);
  __syncthreads();

  const float w = is_q ? q_norm_w[t] : k_norm_w[t];
  const float y = w * x * rsqrtf(var + EPS_F);

  const int   p    = pos_ids[s];
  const float j    = (float)(t & 63);
  const float invf = __expf(-(j * (1.0f / 64.0f)) * LN_ROPE);
  const float fr   = (float)p * invf;
  const float c  = cosf(fr);
  const float sn = sinf(fr);

  red[t] = y;
  __syncthreads();
  const float part = red[t ^ 64];
  const float rot  = (t < 64) ? -part : part;
  float outv = y * c + rot * sn;

  if (is_q) {
    outv *= QSCALE;
    q_proc[(((size_t)b * H_DIM + hh) * S_DIM + s) * D_DIM + t] = outv;
  } else {
    const int kh = hh - 16;
    k_out[(((size_t)b * HKV_DIM + kh) * S_DIM + s) * D_DIM + t] = outv;
  }
}

// ---------------------------------------------------------------------------
// Kernel 3: causal flash attention (GQA), fp32 WMMA + TDM K/V staging.
//   Block = 4 waves, all on the same (b,h), adjacent 16-query tiles ->
//   shared K/V tiles amortized 4x. Wave 0 runs the TDM double-buffer one
//   key-tile ahead; waves past their causal range skip compute but keep the
//   two block barriers uniform. KV_STRIDE=140 (TDM pad 128+12 DW) is
//   conflict-free for both the K-frag (12*i mod 64 distinct) and V-frag
//   (halves 24 banks apart) read patterns.
// ---------------------------------------------------------------------------
#define FA_WAVES 4
#define KV_STRIDE 140   // 128 data + 12 pad DWORDs (TDM pad codes 6/11)

__global__ __launch_bounds__(32 * FA_WAVES)
void flash_attn_wmma(const float* __restrict__ Qp,   // [B,H,S,D], pre-scaled
                     const float* __restrict__ Kp,   // [B,HKV,S,D]
                     const float* __restrict__ Vp,   // [B,HKV,S,D]
                     float* __restrict__ attn) {     // [B,S,H*D]
  __shared__ __align__(16) float Klds[2][16 * KV_STRIDE];
  __shared__ __align__(16) float Vlds[2][16 * KV_STRIDE];
  __shared__ __align__(16) float Plds[FA_WAVES][16 * 17];

  const int NBQ = S_DIM / 16 / FA_WAVES;  // q-tile groups per (b,h)
  const int blk = blockIdx.x;
  const int qg  = blk % NBQ;
  const int h   = (blk / NBQ) % H_DIM;
  const int b   = blk / (NBQ * H_DIM);
  const int kh  = h / G_DIM;

  const int tid  = threadIdx.x;
  const int wave = tid >> 5;
  const int lane = tid & 31;
  const int half = lane >> 4;
  const int lm   = lane & 15;
  const int q0   = (qg * FA_WAVES + wave) * 16;     // this wave's query tile
  const int ntiles = qg * FA_WAVES + FA_WAVES;      // key tiles for the block

  const float* Qrow = Qp + (((size_t)b * H_DIM + h) * S_DIM + (q0 + lm)) * D_DIM;
  v2f qf[32];
#pragma unroll
  for (int i = 0; i < 32; ++i) {
    qf[i].x = Qrow[4 * i + 2 * half];
    qf[i].y = Qrow[4 * i + 2 * half + 1];
  }

  float m8[8], l8[8];
  v8f acc[8];
#pragma unroll
  for (int r = 0; r < 8; ++r) { m8[r] = -1e30f; l8[r] = 0.0f; }
#pragma unroll
  for (int ni = 0; ni < 8; ++ni)
    acc[ni] = (v8f){0.f, 0.f, 0.f, 0.f, 0.f, 0.f, 0.f, 0.f};

  const float* Kbase = Kp + ((size_t)b * HKV_DIM + kh) * (size_t)S_DIM * D_DIM;
  const float* Vbase = Vp + ((size_t)b * HKV_DIM + kh) * (size_t)S_DIM * D_DIM;
  float* Pw = &Plds[wave][0];

  if (wave == 0) {
    tdm_load_2d((u32)(uintptr_t)&Klds[0][0], Kbase, D_DIM, 16, D_DIM, 6u, 11u);
    tdm_load_2d((u32)(uintptr_t)&Vlds[0][0], Vbase, D_DIM, 16, D_DIM, 6u, 11u);
  }

  for (int t = 0; t < ntiles; ++t) {
    const int kb  = t * 16;
    const int buf = t & 1;

    if (wave == 0) {
      if (t + 1 < ntiles) {
        tdm_load_2d((u32)(uintptr_t)&Klds[buf ^ 1][0],
                    Kbase + (size_t)(kb + 16) * D_DIM, D_DIM, 16, D_DIM, 6u, 11u);
        tdm_load_2d((u32)(uintptr_t)&Vlds[buf ^ 1][0],
                    Vbase + (size_t)(kb + 16) * D_DIM, D_DIM, 16, D_DIM, 6u, 11u);
        __builtin_amdgcn_s_wait_tensorcnt(2);
      } else {
        __builtin_amdgcn_s_wait_tensorcnt(0);
      }
    }
    __syncthreads();  // K/V tile [buf] visible to all waves

    if (kb <= q0) {   // causal: this wave still has keys in range
      // ---- scores S = Q * K^T from LDS
      v8f sc = (v8f){0.f, 0.f, 0.f, 0.f, 0.f, 0.f, 0.f, 0.f};
      const float* Kl = &Klds[buf][0];
#pragma unroll
      for (int i = 0; i < 32; ++i) {
        v2f bf;
        bf.x = Kl[lm * KV_STRIDE + 4 * i + 2 * half];
        bf.y = Kl[lm * KV_STRIDE + 4 * i + 2 * half + 1];
        sc = wmma_f32(qf[i], bf, sc);
      }

      if (kb == q0) {  // diagonal tile mask
        const int kg = kb + lm;
#pragma unroll
        for (int r = 0; r < 8; ++r) {
          const int qg2 = q0 + 8 * half + r;
          if (kg > qg2) sc[r] = -1e30f;
        }
      }

      // ---- online softmax (row reductions across each 16-lane half)
      float alpha[8];
#pragma unroll
      for (int r = 0; r < 8; ++r) {
        float rm = sc[r];
        rm = fmaxf(rm, __shfl_xor(rm, 1, 32));
        rm = fmaxf(rm, __shfl_xor(rm, 2, 32));
        rm = fmaxf(rm, __shfl_xor(rm, 4, 32));
        rm = fmaxf(rm, __shfl_xor(rm, 8, 32));
        const float mnew = fmaxf(m8[r], rm);
        const float a    = __expf(m8[r] - mnew);
        const float pv   = __expf(sc[r] - mnew);
        float rs = pv;
        rs += __shfl_xor(rs, 1, 32);
        rs += __shfl_xor(rs, 2, 32);
        rs += __shfl_xor(rs, 4, 32);
        rs += __shfl_xor(rs, 8, 32);
        l8[r]    = a * l8[r] + rs;
        m8[r]    = mnew;
        alpha[r] = a;
        Pw[(8 * half + r) * 17 + lm] = pv;  // per-wave private transpose tile
      }
      // per-wave LDS RAW: hardware LDS pipe is in-order; stop compiler reorder
      asm volatile("" ::: "memory");

#pragma unroll
      for (int ni = 0; ni < 8; ++ni)
#pragma unroll
        for (int r = 0; r < 8; ++r) acc[ni][r] *= alpha[r];

      v2f pf[4];
#pragma unroll
      for (int k4 = 0; k4 < 4; ++k4) {
        pf[k4].x = Pw[lm * 17 + 4 * k4 + 2 * half];
        pf[k4].y = Pw[lm * 17 + 4 * k4 + 2 * half + 1];
      }
      asm volatile("" ::: "memory");

      // ---- O += P * V from LDS
      const float* Vl = &Vlds[buf][0];
#pragma unroll
      for (int ni = 0; ni < 8; ++ni) {
        const int n0 = 16 * ni;
#pragma unroll
        for (int k4 = 0; k4 < 4; ++k4) {
          v2f bf;
          bf.x = Vl[(4 * k4 + 2 * half) * KV_STRIDE + n0 + lm];
          bf.y = Vl[(4 * k4 + 2 * half + 1) * KV_STRIDE + n0 + lm];
          acc[ni] = wmma_f32(pf[k4], bf, acc[ni]);
        }
      }
    }
    __syncthreads();  // all waves done with buf before TDM refills it
  }

  // ---- epilogue: normalize and write [B,S,H*D] for the Wo GEMM
#pragma unroll
  for (int r = 0; r < 8; ++r) {
    const int qg2 = q0 + 8 * half + r;
    const float inv = 1.0f / l8[r];
    float* orow = attn + ((size_t)b * S_DIM + qg2) * (size_t)(H_DIM * D_DIM) + h * D_DIM;
#pragma unroll
    for (int ni = 0; ni < 8; ++ni) orow[16 * ni + lm] = acc[ni][r] * inv;
  }
}

// ---------------------------------------------------------------------------
// Host-side launch
// ---------------------------------------------------------------------------
extern "C" void kernel_launch(void* const* d_in, const int* in_sizes, int n_in,
                              void* d_out, int out_size, void* d_ws, size_t ws_size,
                              hipStream_t stream) {
  const float* hidden  = (const float*)d_in[0];   // [B,S,HID]
  const int*   pos_ids = (const int*)d_in[1];     // [S]
  const float* Wq      = (const float*)d_in[2];   // [HID, H*D]
  const float* Wk      = (const float*)d_in[3];   // [HID, HKV*D]
  const float* Wv      = (const float*)d_in[4];   // [HID, HKV*D]
  const float* Wo      = (const float*)d_in[5];   // [H*D, HID]
  const float* qn      = (const float*)d_in[6];   // [D]
  const float* kn      = (const float*)d_in[7];   // [D]

  float* out   = (float*)d_out;                               // [B,S,HID]
  float* k_out = out + (size_t)B_DIM * S_DIM * HID_DIM;       // [B,HKV,S,D]
  float* v_out = k_out + (size_t)B_DIM * HKV_DIM * S_DIM * D_DIM;

  float* ws     = (float*)d_ws;
  const size_t QRAW = (size_t)B_DIM * S_DIM * H_DIM * D_DIM;    // 8,388,608
  const size_t KRAW = (size_t)B_DIM * S_DIM * HKV_DIM * D_DIM;  // 4,194,304
  float* q_raw  = ws;
  float* k_raw  = q_raw + QRAW;
  float* v_raw  = k_raw + KRAW;
  float* q_proc = v_raw + KRAW;
  float* attn   = q_raw;  // q_raw is dead once q_proc is built

  const int M = B_DIM * S_DIM;  // 4096

  // 1) QKV projections
  gemm_f32_wmma<<<dim3((H_DIM * D_DIM) / GEMM_BN, M / GEMM_BM), 256, 0, stream>>>(
      hidden, Wq, q_raw, M, H_DIM * D_DIM, HID_DIM);
  gemm_f32_wmma<<<dim3((HKV_DIM * D_DIM) / GEMM_BN, M / GEMM_BM), 256, 0, stream>>>(
      hidden, Wk, k_raw, M, HKV_DIM * D_DIM, HID_DIM);
  gemm_f32_wmma<<<dim3((HKV_DIM * D_DIM) / GEMM_BN, M / GEMM_BM), 256, 0, stream>>>(
      hidden, Wv, v_raw, M, HKV_DIM * D_DIM, HID_DIM);

  // 2) RMSNorm + RoPE + reorder (writes k_out/v_out output sections directly)
  norm_rope_reorder<<<B_DIM * S_DIM * 32, 128, 0, stream>>>(
      q_raw, k_raw, v_raw, qn, kn, pos_ids, q_proc, k_out, v_out);

  // 3) causal GQA flash attention (4 q-tile waves share TDM-staged K/V tiles)
  flash_attn_wmma<<<B_DIM * H_DIM * (S_DIM / 16 / FA_WAVES), 32 * FA_WAVES, 0, stream>>>(
      q_proc, k_out, v_out, attn);

  // 4) output projection
  gemm_f32_wmma<<<dim3(HID_DIM / GEMM_BN, M / GEMM_BM), 256, 0, stream>>>(
      attn, Wo, out, M, HID_DIM, H_DIM * D_DIM);
}